// KAN_Convolutional_Layer_31447750542030
// MI455X (gfx1250) — compile-verified
//
#include <hip/hip_runtime.h>

// CDNA5 / gfx1250, wave32. Uses V_WMMA_F32_16X16X4_F32 (full f32 precision),
// one WMMA per 3x3 tap (K=4 features: silu, t, t^2, t^3), 9 accumulating
// WMMAs per 16-pixel tile. M=16 pixels, N=8 convs (cols 8..15 are padding).

typedef float v2f __attribute__((ext_vector_type(2)));
typedef float v4f __attribute__((ext_vector_type(4)));
typedef float v8f __attribute__((ext_vector_type(8)));

#define NB   16   // batch
#define NC   16   // input channels
#define NH   128
#define NW   128
#define NCV  8    // n_convs
#define NTAP 9    // 3x3

// ---------------------------------------------------------------------------
// Prep: fold cheby_coeffs*spline_scaler into 4 per-tap feature weights + bias.
//   features per tap: f0=silu(x), f1=t, f2=t^2, f3=t^3  (t = tanh(x))
//   T0=1, T1=t, T2=2t^2-1, T3=4t^3-3t  =>
//   coef(f0)=base_w, coef(f1)=w1-3w3, coef(f2)=2w2, coef(f3)=4w3,
//   bias_j = sum_i (w0 - w2)
// Wt layout: [tap i][k 0..3][conv j 0..7]  (288 floats), bias: [8]
// ---------------------------------------------------------------------------
__global__ void kan_prep_weights(const float* __restrict__ cheby,
                                 const float* __restrict__ basew,
                                 const float* __restrict__ scal,
                                 float* __restrict__ Wt,
                                 float* __restrict__ bias) {
    int tid = threadIdx.x;              // 0..71
    int i = tid >> 3;                   // tap 0..8
    int j = tid & 7;                    // conv 0..7
    if (i >= NTAP) return;
    float s  = scal[j * NTAP + i];
    const float* c = cheby + (j * NTAP + i) * 4;
    float w1 = c[1] * s, w2 = c[2] * s, w3 = c[3] * s;
    Wt[(i * 4 + 0) * NCV + j] = basew[j * NTAP + i];
    Wt[(i * 4 + 1) * NCV + j] = w1 - 3.0f * w3;
    Wt[(i * 4 + 2) * NCV + j] = 2.0f * w2;
    Wt[(i * 4 + 3) * NCV + j] = 4.0f * w3;
    if (i == 0) {                       // one thread per conv sums the bias
        float bsum = 0.0f;
        for (int ii = 0; ii < NTAP; ++ii) {
            float ss = scal[j * NTAP + ii];
            const float* cc = cheby + (j * NTAP + ii) * 4;
            bsum += (cc[0] - cc[2]) * ss;
        }
        bias[j] = bsum;
    }
}

// ---------------------------------------------------------------------------
// Main kernel: one wave32 per 16-pixel tile of one (b, c, y) row.
// A-frag (16x4 f32): lane m   (m<16) holds {silu, t}   for pixel m (K=0,1)
//                    lane m+16       holds {t^2, t^3}  for pixel m (K=2,3)
// B-frag (4x16 f32): VGPR0 = row K=0 (lanes 0-15) / K=2 (lanes 16-31)
//                    VGPR1 = row K=1 / K=3 ; column = conv j = lane%16
// D (16x16 f32): VGPR r = pixel x0+r (lanes 0-15) / x0+r+8 (lanes 16-31),
//                column = conv j  -> each lane j<8 stores 8 contiguous floats.
// ---------------------------------------------------------------------------
__global__ void __launch_bounds__(256)
kan_conv_wmma(const float* __restrict__ x,
              const float* __restrict__ Wt,
              const float* __restrict__ bias,
              float* __restrict__ out) {
    const int lane = threadIdx.x & 31;
    const int wave = threadIdx.x >> 5;
    const int gw   = blockIdx.x * 8 + wave;     // global wave id, 0..262143

    const int tx = gw & 7;                      // x tile (16 px each)
    const int y  = (gw >> 3) & (NH - 1);
    const int ci = (gw >> 10) & (NC - 1);
    const int bi = gw >> 14;

    const int  m  = lane & 15;                  // pixel-in-tile / conv column
    const bool hi = lane >= 16;
    const int  xo = tx * 16 + m;                // this lane's pixel column

    const float* __restrict__ xin = x + ((size_t)(bi * NC + ci)) * (NH * NW);

    // --- per-wave B fragments (tiny L1-resident table) ---
    const int jj = m & 7;                       // conv index (cols 8..15 dup 0..7)
    const float biasv = bias[jj];
    const int k0 = hi ? 2 : 0;
    const int k1 = hi ? 3 : 1;
    float b0[NTAP], b1[NTAP];
#pragma unroll
    for (int i = 0; i < NTAP; ++i) {
        b0[i] = Wt[(i * 4 + k0) * NCV + jj];
        b1[i] = Wt[(i * 4 + k1) * NCV + jj];
    }

    v8f acc = {};                               // 16x16 f32 accumulator

#pragma unroll
    for (int dy = 0; dy < 3; ++dy) {
        const int  iy  = y + dy - 1;
        const bool yin = (unsigned)iy < (unsigned)NH;
        const float* __restrict__ rowp = xin + (yin ? iy : 0) * NW;
#pragma unroll
        for (int dx = 0; dx < 3; ++dx) {
            const int  ix  = xo + dx - 1;
            const bool inb = yin && ((unsigned)ix < (unsigned)NW);
            const int  cx  = ix < 0 ? 0 : (ix > NW - 1 ? NW - 1 : ix);
            float v = rowp[cx];
            v = inb ? v : 0.0f;                 // zero-pad, EXEC stays full

            // silu(v) = v / (1 + e^-v); tanh(v) = 1 - 2/(e^{2v}+1)
            const float silu = v * (1.0f / (1.0f + __expf(-v)));
            const float e2   = __expf(2.0f * v);
            const float t    = 1.0f - 2.0f / (e2 + 1.0f);
            const float t2   = t * t;

            v2f a;                              // A fragment for this tap
            a.x = hi ? t2 : silu;               // K = 0 / 2
            a.y = hi ? t2 * t : t;              // K = 1 / 3

            const int tap = dy * 3 + dx;
            v2f bf;                             // B fragment for this tap
            bf.x = b0[tap];
            bf.y = b1[tap];

            acc = __builtin_amdgcn_wmma_f32_16x16x4_f32(
                false, a, false, bf, (short)0, acc, false, false);
        }
    }

#pragma unroll
    for (int r = 0; r < 8; ++r) acc[r] += biasv;

    if (m < NCV) {                              // lanes with a real conv column
        const size_t off =
            ((((size_t)bi * NC + ci) * NCV + m) * NH + y) * NW +
            tx * 16 + (hi ? 8 : 0);
        v4f s0 = {acc[0], acc[1], acc[2], acc[3]};
        v4f s1 = {acc[4], acc[5], acc[6], acc[7]};
        __builtin_nontemporal_store(s0, (v4f*)(out + off));      // streamed,
        __builtin_nontemporal_store(s1, (v4f*)(out + off + 4));  // never re-read
    }
}

extern "C" void kernel_launch(void* const* d_in, const int* in_sizes, int n_in,
                              void* d_out, int out_size, void* d_ws, size_t ws_size,
                              hipStream_t stream) {
    const float* x     = (const float*)d_in[0];  // (16,16,128,128)
    const float* cheby = (const float*)d_in[1];  // (8,9,4)
    const float* basew = (const float*)d_in[2];  // (8,9)
    const float* scal  = (const float*)d_in[3];  // (8,9)

    float* Wt   = (float*)d_ws;                  // 288 floats
    float* bias = Wt + NTAP * 4 * NCV;           // 8 floats

    kan_prep_weights<<<1, 72, 0, stream>>>(cheby, basew, scal, Wt, bias);

    // waves = B * C * H * (W/16) = 16*16*128*8 = 262144 ; 8 waves per block
    const int nblocks = (NB * NC * NH * (NW / 16)) / 8;   // 32768
    kan_conv_wmma<<<nblocks, 256, 0, stream>>>(x, Wt, bias, (float*)d_out);
}